// corrs8_893353197605
// MI455X (gfx1250) — compile-verified
//
#include <hip/hip_runtime.h>
#include <hip/hip_bf16.h>
#include <math.h>

// ---------------------------------------------------------------------------
// corrs8 for MI455X (gfx1250, wave32).
//   1) prep_x      : x -> group-channel-last xt (pix*32) + group norms xn
//   2) conv_wmma   : 6 dilated 3x3 convs as implicit GEMM with
//                    V_WMMA_F32_16X16X4_F32, 2x2 register blocking per wave
//                    (32 px x 32 out-ch), reflect padding folded into A
//                    addressing, 32x1152 weight tile in LDS
//   3) yn_kernel   : group norms of conv outputs
//   4) corr_kernel : 81-shift correlation, 24x24x32 y-halo staged in LDS via
//                    GLOBAL_LOAD_ASYNC_TO_LDS_B128 (ASYNCcnt) when available
// ---------------------------------------------------------------------------

typedef __attribute__((ext_vector_type(2))) float v2f;
typedef __attribute__((ext_vector_type(8))) float v8f;
typedef __attribute__((ext_vector_type(4))) int   v4i;

#define H0   96
#define W0   128
#define HW0  (H0 * W0)          // 12288
#define H2   104
#define W2   136
#define HW2  (H2 * W2)          // 14144
#define CIN  128
#define COUT 128
#define KTOT (CIN * 9)          // 1152
#define WSTR 1153               // LDS stride: 1153 % 64 == 1 -> conflict-free
#define YSTR 36                 // corr LDS stride: 16B aligned, conflict-free

#if defined(__gfx1250__) && \
    __has_builtin(__builtin_amdgcn_global_load_async_to_lds_b128) && \
    __has_builtin(__builtin_amdgcn_s_wait_asynccnt)
#define USE_ASYNC_LDS 1
#else
#define USE_ASYNC_LDS 0
#endif

__device__ __forceinline__ int reflect_idx(int m, int n) {
  m = m < 0 ? -m : m;
  m = m >= n ? 2 * n - 2 - m : m;
  return m;
}

// ---------------------------------------------------------------------------
__global__ void prep_x_kernel(const float* __restrict__ x,
                              float* __restrict__ xt,
                              float* __restrict__ xn) {
  int idx = blockIdx.x * blockDim.x + threadIdx.x;   // 2*4*HW0 = 98304
  if (idx >= 2 * 4 * HW0) return;
  int p  = idx % HW0;
  int g  = (idx / HW0) & 3;
  int bb = idx / (4 * HW0);
  const float* src = x + ((size_t)bb * CIN + g * 32) * HW0 + p;
  float* dst = xt + (size_t)idx * 32;
  float s = 0.f;
#pragma unroll
  for (int c = 0; c < 32; ++c) {
    float v = src[(size_t)c * HW0];
    s += v * v;
    dst[c] = v;
  }
  xn[idx] = sqrtf(s);
}

// ---------------------------------------------------------------------------
// grid = (111 pixel-tiles of 128, 4 out-chan groups of 32, b*6 branches)
// block = 128 threads = 4 waves; each wave: 32 px x 32 out-ch (4 WMMA accs).
// ---------------------------------------------------------------------------
__global__ void conv_wmma_kernel(const float* __restrict__ y,
                                 const float* __restrict__ w,
                                 const float* __restrict__ bias,
                                 float* __restrict__ yo) {
  extern __shared__ float ldsW[];                    // 32 * WSTR floats
  const int cg = blockIdx.y;                         // 32-out-channel group
  const int bb = blockIdx.z / 6;
  const int br = blockIdx.z % 6;
  const int d = (br == 0) ? 1 : (br == 1) ? 3 : (br == 2) ? 5
              : (br == 3) ? 9 : (br == 4) ? 13 : 21;

  // stage weights for the 32 output channels of this block (~144KB LDS)
  const float* wsrc = w + ((size_t)br * COUT + cg * 32) * KTOT;
  for (int t = threadIdx.x; t < 32 * KTOT; t += blockDim.x) {
    int n = t / KTOT, kk = t - n * KTOT;
    ldsW[n * WSTR + kk] = wsrc[(size_t)n * KTOT + kk];
  }
  __syncthreads();

  const int lane = threadIdx.x & 31;
  const int wave = threadIdx.x >> 5;
  const int row  = lane & 15;            // A row (pixel) / B,D col (channel)
  const int koff = (lane >> 4) << 1;     // lanes 16-31 carry K = 2,3

  const int p0   = blockIdx.x * 128 + wave * 32;
  const int pixA = p0 + row;
  const int pixB = p0 + 16 + row;
  const int pyA = pixA / W2, pxA = pixA % W2;
  const int pyB = pixB / W2, pxB = pixB % W2;
  const bool inA = pixA < HW2, inB = pixB < HW2;
  const float* ybase = y + (size_t)bb * CIN * HW0;

  v8f acc00 = {}, acc01 = {}, acc10 = {}, acc11 = {};
  const int wb0 = row * WSTR;            // B cols cg*32 + 0..15
  const int wb1 = (row + 16) * WSTR;     // B cols cg*32 + 16..31

#pragma unroll
  for (int tap = 0; tap < 9; ++tap) {
    const int dy = (tap / 3 - 1) * d;
    const int dx = (tap % 3 - 1) * d;

    const int ryA = pyA + dy, rxA = pxA + dx;
    const bool validA = inA & (ryA >= 0) & (ryA < H2) & (rxA >= 0) & (rxA < W2);
    const float* srcA =
        ybase + reflect_idx(ryA - 4, H0) * W0 + reflect_idx(rxA - 4, W0);

    const int ryB = pyB + dy, rxB = pxB + dx;
    const bool validB = inB & (ryB >= 0) & (ryB < H2) & (rxB >= 0) & (rxB < W2);
    const float* srcB =
        ybase + reflect_idx(ryB - 4, H0) * W0 + reflect_idx(rxB - 4, W0);

#pragma unroll 4
    for (int c0 = 0; c0 < CIN; c0 += 4) {
      const int c = c0 + koff;
      v2f a0, a1, bf0, bf1;
      a0.x = validA ? srcA[(size_t)c * HW0]       : 0.f;
      a0.y = validA ? srcA[(size_t)(c + 1) * HW0] : 0.f;
      a1.x = validB ? srcB[(size_t)c * HW0]       : 0.f;
      a1.y = validB ? srcB[(size_t)(c + 1) * HW0] : 0.f;
      bf0.x = ldsW[wb0 + c * 9 + tap];
      bf0.y = ldsW[wb0 + (c + 1) * 9 + tap];
      bf1.x = ldsW[wb1 + c * 9 + tap];
      bf1.y = ldsW[wb1 + (c + 1) * 9 + tap];
      acc00 = __builtin_amdgcn_wmma_f32_16x16x4_f32(false, a0, false, bf0,
                                                    (short)0, acc00, false, false);
      acc01 = __builtin_amdgcn_wmma_f32_16x16x4_f32(false, a0, false, bf1,
                                                    (short)0, acc01, false, false);
      acc10 = __builtin_amdgcn_wmma_f32_16x16x4_f32(false, a1, false, bf0,
                                                    (short)0, acc10, false, false);
      acc11 = __builtin_amdgcn_wmma_f32_16x16x4_f32(false, a1, false, bf1,
                                                    (short)0, acc11, false, false);
    }
  }

  // epilogue: +bias, channel-last per group (group == cg since N-tile is 32)
  const int n = row;
  const float bv0 = bias[br * COUT + cg * 32 + n];
  const float bv1 = bias[br * COUT + cg * 32 + 16 + n];
  float* obase = yo + ((size_t)((bb * 6 + br) * 4 + cg) * HW2) * 32;
  const int rb0 = p0 + ((lane >> 4) << 3);
  const int rb1 = p0 + 16 + ((lane >> 4) << 3);
#pragma unroll
  for (int r = 0; r < 8; ++r) {
    if (rb0 + r < HW2) {
      obase[(size_t)(rb0 + r) * 32 + n]      = acc00[r] + bv0;
      obase[(size_t)(rb0 + r) * 32 + n + 16] = acc01[r] + bv1;
    }
    if (rb1 + r < HW2) {
      obase[(size_t)(rb1 + r) * 32 + n]      = acc10[r] + bv0;
      obase[(size_t)(rb1 + r) * 32 + n + 16] = acc11[r] + bv1;
    }
  }
}

// ---------------------------------------------------------------------------
__global__ void yn_kernel(const float* __restrict__ yo,
                          float* __restrict__ yn) {
  int idx = blockIdx.x * blockDim.x + threadIdx.x;   // 2*6*4*HW2 = 678912
  if (idx >= 2 * 6 * 4 * HW2) return;
  const float4* src = (const float4*)(yo + (size_t)idx * 32);
  float s = 0.f;
#pragma unroll
  for (int q = 0; q < 8; ++q) {
    float4 v = src[q];
    s += v.x * v.x + v.y * v.y + v.z * v.z + v.w * v.w;
  }
  yn[idx] = sqrtf(s);
}

// ---------------------------------------------------------------------------
__global__ void corr_kernel(const float* __restrict__ yo,
                            const float* __restrict__ yn,
                            const float* __restrict__ xt,
                            const float* __restrict__ xn,
                            float* __restrict__ out) {
  extern __shared__ float lds[];
  float* ldsY = lds;                 // 576 * YSTR
  float* ldsN = lds + 576 * YSTR;    // 576
  const int tile = blockIdx.x;       // 6 x 8 tiles of 16x16 pixels
  const int th = (tile >> 3) * 16;
  const int tw = (tile & 7) * 16;
  const int z  = blockIdx.y;         // bb*24 + br*4 + g
  const int bb = z / 24;
  const int br = (z / 4) % 6;
  const int g  = z & 3;

  const size_t ybase = (size_t)((bb * 6 + br) * 4 + g) * HW2;

  // stage y halo (24x24 pixels x 32ch)
#if USE_ASYNC_LDS
  for (int i = threadIdx.x; i < 576 * 8; i += blockDim.x) {
    int pl = i >> 3, q = i & 7;
    int rr = pl / 24, cw = pl - rr * 24;
    size_t gp = ybase + (size_t)(th + rr) * W2 + (tw + cw);
    const float* src = yo + gp * 32 + q * 4;
    float* dst = ldsY + pl * YSTR + q * 4;
    __builtin_amdgcn_global_load_async_to_lds_b128(
        (__attribute__((address_space(1))) v4i*)src,
        (__attribute__((address_space(3))) v4i*)dst, 0, 0);
  }
  __builtin_amdgcn_s_wait_asynccnt(0);
#else
  const float4* ysrc = (const float4*)yo;
  for (int i = threadIdx.x; i < 576 * 8; i += blockDim.x) {
    int pl = i >> 3, q = i & 7;
    int rr = pl / 24, cw = pl - rr * 24;
    size_t gp = ybase + (size_t)(th + rr) * W2 + (tw + cw);
    *(float4*)(ldsY + pl * YSTR + q * 4) = ysrc[gp * 8 + q];
  }
#endif
  for (int i = threadIdx.x; i < 576; i += blockDim.x) {
    int rr = i / 24, cw = i - rr * 24;
    ldsN[i] = yn[ybase + (size_t)(th + rr) * W2 + (tw + cw)];
  }
  __syncthreads();

  const int r  = threadIdx.x >> 4;
  const int cc = threadIdx.x & 15;
  const int hh = th + r, ww = tw + cc;
  const int p  = hh * W0 + ww;

  float xv[32];
  const size_t xbase = ((size_t)(bb * 4 + g) * HW0 + p) * 32;
#pragma unroll
  for (int q = 0; q < 8; ++q) {
    float4 v = *(const float4*)(xt + xbase + q * 4);
    xv[q * 4 + 0] = v.x; xv[q * 4 + 1] = v.y;
    xv[q * 4 + 2] = v.z; xv[q * 4 + 3] = v.w;
  }
  const float xnv = xn[(size_t)(bb * 4 + g) * HW0 + p];

  const int c24 = br * 4 + g;
  float* obase = out + (size_t)(bb * 24 + c24) * 81 * HW0 + p;
  for (int s = 0; s < 81; ++s) {
    int sx = s / 9, sy = s - sx * 9;
    const int hp = (r + sx) * 24 + (cc + sy);
    const float* yv = ldsY + hp * YSTR;
    float dot = 0.f;
#pragma unroll
    for (int c = 0; c < 32; ++c) dot += xv[c] * yv[c];
    obase[(size_t)s * HW0] = dot / fmaxf(xnv * ldsN[hp], 1e-8f);
  }
}

// ---------------------------------------------------------------------------
extern "C" void kernel_launch(void* const* d_in, const int* in_sizes, int n_in,
                              void* d_out, int out_size, void* d_ws, size_t ws_size,
                              hipStream_t stream) {
  const float* x    = (const float*)d_in[0];   // (2,128,96,128)
  const float* y    = (const float*)d_in[1];   // (2,128,96,128)
  const float* w    = (const float*)d_in[2];   // (6,128,128,3,3)
  const float* bias = (const float*)d_in[3];   // (6,128)
  float* out = (float*)d_out;                  // (2,24,81,96,128)

  float* yo = (float*)d_ws;                            // 2*6*4*HW2*32
  float* yn = yo + (size_t)2 * 6 * 4 * HW2 * 32;       // 2*6*4*HW2
  float* xt = yn + (size_t)2 * 6 * 4 * HW2;            // 2*4*HW0*32
  float* xn = xt + (size_t)2 * 4 * HW0 * 32;           // 2*4*HW0

  const size_t convLds = (size_t)32 * WSTR * sizeof(float);          // ~144KB
  const size_t corrLds = (size_t)(576 * YSTR + 576) * sizeof(float); // ~83KB
  (void)hipFuncSetAttribute((const void*)conv_wmma_kernel,
                            hipFuncAttributeMaxDynamicSharedMemorySize,
                            (int)convLds);
  (void)hipFuncSetAttribute((const void*)corr_kernel,
                            hipFuncAttributeMaxDynamicSharedMemorySize,
                            (int)corrLds);

  prep_x_kernel<<<(2 * 4 * HW0 + 255) / 256, 256, 0, stream>>>(x, xt, xn);
  conv_wmma_kernel<<<dim3((HW2 + 127) / 128, COUT / 32, 2 * 6), 128, convLds,
                     stream>>>(y, w, bias, yo);
  yn_kernel<<<(2 * 6 * 4 * HW2 + 255) / 256, 256, 0, stream>>>(yo, yn);
  corr_kernel<<<dim3(48, 48), 256, corrLds, stream>>>(yo, yn, xt, xn, out);
}